// VoxelHashMap_91336774517475
// MI455X (gfx1250) — compile-verified
//
#include <hip/hip_runtime.h>
#include <hip/hip_bf16.h>

typedef float f4 __attribute__((ext_vector_type(4)));

#define KPV 8            // points per voxel (ring buffer depth)
#define VRES 0.1f

// ---------------------------------------------------------------------------
// Pass-through copy, non-temporal 128-bit path (don't pollute 192MB L2 with a
// 660MB stream that is never re-read).
// ---------------------------------------------------------------------------
__global__ void k_copy4_nt(const f4* __restrict__ src, f4* __restrict__ dst, long n4) {
  long i  = (long)blockIdx.x * blockDim.x + threadIdx.x;
  long st = (long)gridDim.x * blockDim.x;
  for (; i < n4; i += st) {
    f4 v = __builtin_nontemporal_load(&src[i]);
    __builtin_nontemporal_store(v, &dst[i]);
  }
}

// ---------------------------------------------------------------------------
// Pass-through copy of the big descriptor buffer via the CDNA5 async
// global->LDS->global path (ASYNCcnt-tracked, no VGPR staging).
// Each lane owns a private 16B LDS slot. ASYNCcnt cannot tell a completed
// load from a completed store (they retire out-of-order w.r.t. each other),
// so each wave fences to 0 around the LDS slot reuse; memory-level
// parallelism comes from the 8 waves/block x 4096 blocks.
// ---------------------------------------------------------------------------
__global__ void k_copy4_async(const f4* __restrict__ src, f4* __restrict__ dst, long n4) {
  __shared__ f4 stage[256];
  // Generic pointer to LDS: low 32 bits are the LDS byte offset.
  unsigned lds = (unsigned)(size_t)&stage[threadIdx.x];
  long i  = (long)blockIdx.x * blockDim.x + threadIdx.x;
  long st = (long)gridDim.x * blockDim.x;
  for (; i < n4; i += st) {
    const f4* s = src + i;
    f4*       d = dst + i;
    // ensure previous iteration's async store has drained before reusing LDS
    asm volatile("s_wait_asynccnt 0" ::: "memory");
    asm volatile("global_load_async_to_lds_b128 %0, %1, off"
                 :: "v"(lds), "v"(s) : "memory");
    asm volatile("s_wait_asynccnt 0" ::: "memory");
    asm volatile("global_store_async_from_lds_b128 %0, %1, off"
                 :: "v"(d), "v"(lds) : "memory");
  }
  asm volatile("s_wait_asynccnt 0" ::: "memory");
}

// ---------------------------------------------------------------------------
// Scratch init: chain heads = -1
// ---------------------------------------------------------------------------
__global__ void k_init_heads(int* __restrict__ head, int nbuf) {
  int i  = blockIdx.x * blockDim.x + threadIdx.x;
  int st = gridDim.x * blockDim.x;
  for (; i < nbuf; i += st) head[i] = -1;
}

// ---------------------------------------------------------------------------
// Hash each point (matches jnp int32 wraparound + non-negative remainder)
// and push onto its voxel's linked list.
// ---------------------------------------------------------------------------
__global__ void k_hash_link(const float* __restrict__ pts, int n, int nbuf,
                            int* __restrict__ h_arr,
                            int* __restrict__ head, int* __restrict__ nxt) {
  int i = blockIdx.x * blockDim.x + threadIdx.x;
  if (i >= n) return;
  float x = pts[3l*i + 0], y = pts[3l*i + 1], z = pts[3l*i + 2];
  int gx = (int)floorf(x / VRES);
  int gy = (int)floorf(y / VRES);
  int gz = (int)floorf(z / VRES);
  int s  = (int)((unsigned)gx * 73856093u +
                 (unsigned)gy * 19349669u +
                 (unsigned)gz * 83492791u);      // int32 wraparound like jnp
  int h = s % nbuf;
  if (h < 0) h += nbuf;                          // jnp.remainder is non-negative
  h_arr[i] = h;
  nxt[i] = atomicExch(&head[h], i);              // chain order irrelevant (see rank)
}

// ---------------------------------------------------------------------------
// Deterministic rank + scatter. Walking the chain and counting (j < i) gives
// the point's rank in ORIGINAL index order, independent of the
// nondeterministic chain order. Keep the last K of each bucket, ring-insert.
// Streaming payloads (descriptors, ts) use NT hints; the hot chain arrays
// (10MB, re-read by k_newcount) keep default RT residency in L2.
// ---------------------------------------------------------------------------
__global__ void k_scatter(const float* __restrict__ pts,
                          const float* __restrict__ desc,
                          const float* __restrict__ rgb,
                          const int*   __restrict__ vcount,
                          const int*   __restrict__ frame_id,
                          const int*   __restrict__ h_arr,
                          const int*   __restrict__ head,
                          const int*   __restrict__ nxt,
                          int n,
                          float* __restrict__ o_pts,
                          float* __restrict__ o_desc,
                          float* __restrict__ o_rgb,
                          int*   __restrict__ o_tsc,
                          int*   __restrict__ o_tsu,
                          int*   __restrict__ o_st) {
  int i = blockIdx.x * blockDim.x + threadIdx.x;
  if (i >= n) return;
  int fid = frame_id[0];                         // scalar load overlaps chain walk
  int h = h_arr[i];
  int cnt = 0, r = 0;
  for (int j = head[h]; j >= 0; j = nxt[j]) {
    cnt++;
    r += (j < i) ? 1 : 0;
  }
  int cutoff = cnt - KPV;
  if (r < cutoff) return;                        // not among the last K -> dropped
  int relf = r - (cutoff > 0 ? cutoff : 0);      // rank among kept points
  int base = vcount[h];
  int slot = (base + relf) % KPV;                // ring-buffer slot (unique per voxel)
  long b = (long)h * KPV + slot;

  o_pts[3*b + 0] = pts[3l*i + 0];
  o_pts[3*b + 1] = pts[3l*i + 1];
  o_pts[3*b + 2] = pts[3l*i + 2];
  o_rgb[3*b + 0] = rgb[3l*i + 0];
  o_rgb[3*b + 1] = rgb[3l*i + 1];
  o_rgb[3*b + 2] = rgb[3l*i + 2];
  __builtin_nontemporal_store(fid, &o_tsc[b]);
  __builtin_nontemporal_store(fid, &o_tsu[b]);
  __builtin_nontemporal_store(fid, &o_st[b]);
  // 128B descriptor: 8 x b128, read-once / write-once -> NT both ways
  const f4* ds = (const f4*)(desc + 32l * i);
  f4*       dd = (f4*)(o_desc + 32l * b);
#pragma unroll
  for (int q = 0; q < 8; ++q) {
    f4 v = __builtin_nontemporal_load(&ds[q]);
    __builtin_nontemporal_store(v, &dd[q]);
  }
}

// ---------------------------------------------------------------------------
// new_count[v] = old_count[v] + min(bucket_size, K)
// ---------------------------------------------------------------------------
__global__ void k_newcount(const int* __restrict__ vcount,
                           const int* __restrict__ head,
                           const int* __restrict__ nxt,
                           int nbuf, int* __restrict__ o_cnt) {
  int v = blockIdx.x * blockDim.x + threadIdx.x;
  if (v >= nbuf) return;
  int c = 0;
  for (int j = head[v]; j >= 0; j = nxt[j]) c++;
  __builtin_nontemporal_store(vcount[v] + (c < KPV ? c : KPV), &o_cnt[v]);
}

// ---------------------------------------------------------------------------
extern "C" void kernel_launch(void* const* d_in, const int* in_sizes, int n_in,
                              void* d_out, int out_size, void* d_ws, size_t ws_size,
                              hipStream_t stream) {
  const float* points   = (const float*)d_in[0];
  const float* descr    = (const float*)d_in[1];
  const float* rgb      = (const float*)d_in[2];
  const float* bpts     = (const float*)d_in[3];
  const float* bdesc    = (const float*)d_in[4];
  const float* brgb     = (const float*)d_in[5];
  const int*   btsc     = (const int*)d_in[6];
  const int*   btsu     = (const int*)d_in[7];
  const int*   bst      = (const int*)d_in[8];
  const int*   vcount   = (const int*)d_in[9];
  const int*   frame_id = (const int*)d_in[10];

  const int  N    = in_sizes[0] / 3;     // 1,000,000
  const long CAP  = in_sizes[3] / 3;     // 4,000,000
  const int  NBUF = in_sizes[9];         // 500,000

  // output layout: flat concat in reference return order
  float* out    = (float*)d_out;
  float* o_pts  = out;
  float* o_desc = o_pts  + CAP * 3;
  float* o_rgb  = o_desc + CAP * 32;
  int*   o_tsc  = (int*)(o_rgb + CAP * 3);
  int*   o_tsu  = o_tsc + CAP;
  int*   o_st   = o_tsu + CAP;
  int*   o_cnt  = o_st  + CAP;

  // workspace: chain heads [NBUF] | next [N] | hash [N]
  int* ws_head = (int*)d_ws;
  int* ws_next = ws_head + NBUF;
  int* ws_hash = ws_next + N;

  const int T = 256;

  // --- pass-through copies (overwritten at scattered slots afterwards) ---
  // descriptors: 512MB via async global->LDS->global path
  k_copy4_async<<<4096, T, 0, stream>>>((const f4*)bdesc, (f4*)o_desc, CAP * 32 / 4);
  // remaining buffers: NT b128 copies
  k_copy4_nt<<<2048, T, 0, stream>>>((const f4*)bpts, (f4*)o_pts, CAP * 3 / 4);
  k_copy4_nt<<<2048, T, 0, stream>>>((const f4*)brgb, (f4*)o_rgb, CAP * 3 / 4);
  k_copy4_nt<<<1024, T, 0, stream>>>((const f4*)btsc, (f4*)o_tsc, CAP / 4);
  k_copy4_nt<<<1024, T, 0, stream>>>((const f4*)btsu, (f4*)o_tsu, CAP / 4);
  k_copy4_nt<<<1024, T, 0, stream>>>((const f4*)bst,  (f4*)o_st,  CAP / 4);

  // --- build per-voxel chains ---
  k_init_heads<<<(NBUF + T - 1) / T, T, 0, stream>>>(ws_head, NBUF);
  k_hash_link<<<(N + T - 1) / T, T, 0, stream>>>(points, N, NBUF, ws_hash, ws_head, ws_next);

  // --- deterministic rank + ring-buffer scatter ---
  k_scatter<<<(N + T - 1) / T, T, 0, stream>>>(points, descr, rgb, vcount, frame_id,
                                               ws_hash, ws_head, ws_next, N,
                                               o_pts, o_desc, o_rgb, o_tsc, o_tsu, o_st);

  // --- per-voxel insertion counts ---
  k_newcount<<<(NBUF + T - 1) / T, T, 0, stream>>>(vcount, ws_head, ws_next, NBUF, o_cnt);
}